// PhyloATTN_11252814316085
// MI455X (gfx1250) — compile-verified
//
#include <hip/hip_runtime.h>
#include <hip/hip_bf16.h>
#include <math.h>

#define BB 2
#define RR 64
#define CC 32
#define DD 128
#define NN 2016
#define CD 4096              // C*D
#define TILE_N 4             // pairs per workgroup
#define ROWS 128             // TILE_N * CC
#define LSTRIDE 132          // padded LDS row stride (conflict-free stride-128 row reads)
#define BUFE (ROWS * LSTRIDE)
#define SCALE 0.015625f      // 1/sqrt(C*D) = 1/64

typedef float v2f __attribute__((ext_vector_type(2)));
typedef float v4f __attribute__((ext_vector_type(4)));
typedef float v8f __attribute__((ext_vector_type(8)));

// Stage a [128][128] row-major global weight into LDS transposed: WT[n][k], stride LSTRIDE.
__device__ __forceinline__ void stage_WT(const float* __restrict__ W,
                                         float* __restrict__ WT, int tid)
{
#pragma unroll
    for (int e = 0; e < 16; ++e) {
        int i = (tid + 256 * e) << 2;     // 16384 floats, float4 granularity
        int k = i >> 7, n = i & 127;
        v4f w = *(const v4f*)(W + i);
        WT[(n + 0) * LSTRIDE + k] = w.x;
        WT[(n + 1) * LSTRIDE + k] = w.y;
        WT[(n + 2) * LSTRIDE + k] = w.z;
        WT[(n + 3) * LSTRIDE + k] = w.w;
    }
}

// D(128x128, LDS) = (A1[-A2]) @ W + bias, W given transposed in LDS (WT[n][k]).
// 256 threads = 8 wave32. Wave w owns output column block nt=w; loops 8 M-tiles.
// All operands come from LDS as 8-byte conflict-free ds loads.
__device__ __forceinline__ void gemm128(const float* __restrict__ A1,
                                        const float* __restrict__ A2,
                                        const float* __restrict__ WT,
                                        const float* __restrict__ bias,
                                        float* __restrict__ Dout, int tid)
{
    const int lane = tid & 31;
    const int wave = tid >> 5;        // nt = wave
    const int l16  = lane >> 4;
    const int lm   = lane & 15;
    const int ncol = (wave << 4) + lm;
    const float* Wrow = WT + ncol * LSTRIDE;   // WT[ncol][*]
    const float  bv   = bias[ncol];
#pragma unroll 1
    for (int mt = 0; mt < 8; ++mt) {
        const int arow = (mt << 4) + lm;
        const float* Ar1 = A1 + arow * LSTRIDE;
        const float* Ar2 = A2 ? (A2 + arow * LSTRIDE) : nullptr;
        v8f acc = {0.f,0.f,0.f,0.f,0.f,0.f,0.f,0.f};
        for (int k0 = 0; k0 < DD; k0 += 4) {
            const int kb = k0 + (l16 << 1);   // K = 2*(lane/16) + v
            v2f a = *(const v2f*)(Ar1 + kb);
            if (A2) { v2f s = *(const v2f*)(Ar2 + kb); a.x -= s.x; a.y -= s.y; }
            v2f b = *(const v2f*)(Wrow + kb);
            acc = __builtin_amdgcn_wmma_f32_16x16x4_f32(false, a, false, b,
                                                        (short)0, acc, false, false);
        }
#pragma unroll
        for (int v = 0; v < 8; ++v) {
            const int row = (mt << 4) + v + (l16 << 3);
            Dout[row * LSTRIDE + ncol] = acc[v] + bv;
        }
    }
}

// Kernel 1: kproj = batch_input @ Wk + bk ; each block does 4 consecutive r rows.
__global__ __launch_bounds__(256) void kproj_kernel(const float* __restrict__ batch_input,
                                                    const float* __restrict__ Wk,
                                                    const float* __restrict__ bk,
                                                    float* __restrict__ kproj)
{
    extern __shared__ float smem[];
    float* sA  = smem;
    float* sT  = smem + BUFE;
    float* sWT = smem + 2 * BUFE;
    const int tid = threadIdx.x;
    const float* src = batch_input + (size_t)blockIdx.x * (TILE_N * CD);
#pragma unroll
    for (int e = 0; e < 16; ++e) {
        int i = (tid + 256 * e) << 2;
        *(v4f*)(sA + (i >> 7) * LSTRIDE + (i & 127)) = *(const v4f*)(src + i);
    }
    stage_WT(Wk, sWT, tid);
    __syncthreads();
    gemm128(sA, nullptr, sWT, bk, sT, tid);
    __syncthreads();
    float* dst = kproj + (size_t)blockIdx.x * (TILE_N * CD);
#pragma unroll
    for (int e = 0; e < 16; ++e) {
        int i = (tid + 256 * e) << 2;
        *(v4f*)(dst + i) = *(const v4f*)(sT + (i >> 7) * LSTRIDE + (i & 127));
    }
}

// Kernel 2: fused pipeline, 4 pairs per workgroup. grid = B*N/4 = 1008, 256 threads.
__global__ __launch_bounds__(256) void phylo_kernel(
    const float* __restrict__ x_i, const float* __restrict__ x_j,
    const float* __restrict__ batch_input, const unsigned char* __restrict__ seq_mask,
    const int* __restrict__ i_idx, const int* __restrict__ j_idx,
    const float* __restrict__ Wh, const float* __restrict__ bh,
    const float* __restrict__ Wq, const float* __restrict__ bq,
    const float* __restrict__ Wg, const float* __restrict__ bg,
    const float* __restrict__ Ws1, const float* __restrict__ bs1,
    const float* __restrict__ Ws2, const float* __restrict__ bs2,
    const float* __restrict__ kproj, float* __restrict__ out)
{
    extern __shared__ float smem[];
    float* sXi = smem;               // xi -> q -> g_raw -> gelu(t)
    float* sXj = smem + BUFE;        // xj -> x_glob
    float* sT  = smem + 2 * BUFE;    // h  -> x -> xfin
    float* sWT = smem + 3 * BUFE;    // staged (transposed) weight of current GEMM
    __shared__ float sAlpha[TILE_N * RR];
    __shared__ float sParts[256];
    __shared__ float sSc[ROWS];

    const int tid = threadIdx.x;
    const int b   = blockIdx.x / (NN / TILE_N);
    const int n0  = (blockIdx.x % (NN / TILE_N)) * TILE_N;
    const size_t base = (size_t)(b * NN + n0) * CD;   // 4 pairs are contiguous

    // ---- load x_i/x_j tiles (16384 floats each) + stage Wh
#pragma unroll
    for (int e = 0; e < 16; ++e) {
        int i = (tid + 256 * e) << 2;
        int p = (i >> 7) * LSTRIDE + (i & 127);
        *(v4f*)(sXi + p) = *(const v4f*)(x_i + base + i);
        *(v4f*)(sXj + p) = *(const v4f*)(x_j + base + i);
    }
    stage_WT(Wh, sWT, tid);
    __syncthreads();

    // ---- h = (xi - xj) @ Wh + bh -> sT
    gemm128(sXi, sXj, sWT, bh, sT, tid);
    __syncthreads();

    // ---- x = z*xi + (1-z)*xj (overwrite sT) ; stage Wq
    for (int e = 0; e < 64; ++e) {
        int i = tid + 256 * e;
        int p = (i >> 7) * LSTRIDE + (i & 127);
        float z = 1.f / (1.f + expf(-sT[p]));
        sT[p] = z * sXi[p] + (1.f - z) * sXj[p];
    }
    stage_WT(Wq, sWT, tid);
    __syncthreads();

    // ---- q = x @ Wq + bq -> sXi
    gemm128(sT, nullptr, sWT, bq, sXi, tid);
    __syncthreads();

    // ---- alpha_raw[p][r] = <q_p, kproj[b][r]> (one thread per (p,r)) ; stage Wg
    {
        const int p = tid >> 6, r = tid & 63;
        const float* kr = kproj + (size_t)(b * RR + r) * CD;
        const float* qb = sXi + (p * CC) * LSTRIDE;
        float acc = 0.f;
        for (int j = 0; j < CD; j += 4) {
            v4f qv = *(const v4f*)(qb + (j >> 7) * LSTRIDE + (j & 127));
            v4f kv = *(const v4f*)(kr + j);
            acc += qv.x * kv.x + qv.y * kv.y + qv.z * kv.z + qv.w * kv.w;
        }
        sAlpha[p * RR + r] = acc;
    }
    stage_WT(Wg, sWT, tid);
    __syncthreads();

    // ---- scale + mask + softmax (one thread per pair; 64 entries each)
    if (tid < TILE_N) {
        const int n = n0 + tid;
        const int ii = i_idx[n], jj = j_idx[n];
        float* al = sAlpha + tid * RR;
        float m = -__builtin_inff();
        for (int r = 0; r < RR; ++r) {
            float a = al[r] * SCALE;
            if (r == ii || r == jj) a = -__builtin_inff();
            al[r] = a; m = fmaxf(m, a);
        }
        float s = 0.f;
        for (int r = 0; r < RR; ++r) { float e = expf(al[r] - m); al[r] = e; s += e; }
        float inv = 1.f / s;
        for (int r = 0; r < RR; ++r) al[r] *= inv;
    }
    __syncthreads();

    // ---- x_glob = sum_r alpha[r]*batch_input[b][r]; one batch_input read feeds 4 pairs
    for (int e = 0; e < 16; ++e) {
        int cd = tid + 256 * e;               // 0..4095
        const float* bp = batch_input + (size_t)b * RR * CD + cd;
        float a0 = 0.f, a1 = 0.f, a2 = 0.f, a3 = 0.f;
        for (int r = 0; r < RR; ++r) {
            float bi = bp[(size_t)r * CD];
            a0 += sAlpha[0 * RR + r] * bi;
            a1 += sAlpha[1 * RR + r] * bi;
            a2 += sAlpha[2 * RR + r] * bi;
            a3 += sAlpha[3 * RR + r] * bi;
        }
        int p = (cd >> 7) * LSTRIDE + (cd & 127);
        sXj[(0 * CC) * LSTRIDE + p] = a0;
        sXj[(1 * CC) * LSTRIDE + p] = a1;
        sXj[(2 * CC) * LSTRIDE + p] = a2;
        sXj[(3 * CC) * LSTRIDE + p] = a3;
    }
    __syncthreads();

    // ---- g = x_glob @ Wg + bg -> sXi   (q is dead)
    gemm128(sXj, nullptr, sWT, bg, sXi, tid);
    __syncthreads();

    // ---- xfin = (1-w)*x + w*x_glob, w = sigmoid(g) (overwrite sT) ; stage Ws1
    for (int e = 0; e < 64; ++e) {
        int i = tid + 256 * e;
        int p = (i >> 7) * LSTRIDE + (i & 127);
        float w = 1.f / (1.f + expf(-sXi[p]));
        sT[p] = (1.f - w) * sT[p] + w * sXj[p];
    }
    stage_WT(Ws1, sWT, tid);
    __syncthreads();

    // ---- t = xfin @ Ws1 + bs1 -> sXi
    gemm128(sT, nullptr, sWT, bs1, sXi, tid);
    __syncthreads();

    // ---- exact gelu in place
    for (int e = 0; e < 64; ++e) {
        int i = tid + 256 * e;
        int p = (i >> 7) * LSTRIDE + (i & 127);
        float v = sXi[p];
        sXi[p] = 0.5f * v * (1.f + erff(v * 0.70710678118654752f));
    }
    __syncthreads();

    // ---- s[p][c] = gelu_t . Ws2 + bs2 ; score[p] = sum_c s[p][c]*mask[b][c]
    {
        int pc = tid >> 1, seg = tid & 1;     // row index pc = p*32 + c
        const float* row = sXi + pc * LSTRIDE;
        float acc = 0.f;
        for (int d = seg * 64; d < seg * 64 + 64; ++d) acc += row[d] * Ws2[d];
        sParts[tid] = acc;
    }
    __syncthreads();
    if (tid < ROWS) {
        int c = tid & 31;
        float s = sParts[2 * tid] + sParts[2 * tid + 1] + bs2[0];
        sSc[tid] = seq_mask[b * CC + c] ? s : 0.f;
    }
    __syncthreads();
    if (tid < TILE_N) {
        float s = 0.f;
        for (int c = 0; c < CC; ++c) s += sSc[tid * CC + c];
        out[b * NN + n0 + tid] = s;
    }
}

extern "C" void kernel_launch(void* const* d_in, const int* in_sizes, int n_in,
                              void* d_out, int out_size, void* d_ws, size_t ws_size,
                              hipStream_t stream) {
    const float* x_i         = (const float*)d_in[0];
    const float* x_j         = (const float*)d_in[1];
    const float* batch_input = (const float*)d_in[2];
    const unsigned char* seq_mask = (const unsigned char*)d_in[3];
    const int* i_idx = (const int*)d_in[4];
    const int* j_idx = (const int*)d_in[5];
    const float* Wh  = (const float*)d_in[6];  const float* bh  = (const float*)d_in[7];
    const float* Wq  = (const float*)d_in[8];  const float* bq  = (const float*)d_in[9];
    const float* Wk  = (const float*)d_in[10]; const float* bk  = (const float*)d_in[11];
    const float* Wg  = (const float*)d_in[12]; const float* bg  = (const float*)d_in[13];
    const float* Ws1 = (const float*)d_in[14]; const float* bs1 = (const float*)d_in[15];
    const float* Ws2 = (const float*)d_in[16]; const float* bs2 = (const float*)d_in[17];

    float* kproj = (float*)d_ws;      // B*R*C*D floats = 2 MB scratch
    float* out   = (float*)d_out;     // [B, N]

    const size_t lds1 = (size_t)3 * BUFE * sizeof(float);   // ~203 KB
    const size_t lds2 = (size_t)4 * BUFE * sizeof(float);   // ~270 KB (of 320 KB/WGP)

    kproj_kernel<<<dim3(BB * RR / TILE_N), dim3(256), lds1, stream>>>(batch_input, Wk, bk, kproj);
    phylo_kernel<<<dim3(BB * NN / TILE_N), dim3(256), lds2, stream>>>(
        x_i, x_j, batch_input, seq_mask, i_idx, j_idx,
        Wh, bh, Wq, bq, Wg, bg, Ws1, bs1, Ws2, bs2, kproj, out);
}